// Net_19387482374339
// MI455X (gfx1250) — compile-verified
//
#include <hip/hip_runtime.h>

#define AS1 __attribute__((address_space(1)))
#define AS3 __attribute__((address_space(3)))

typedef int v4i __attribute__((ext_vector_type(4)));

// ---- gfx1250 async global->LDS copy (ASYNCcnt path) ------------------------
__device__ __forceinline__ void async_copy_b128(const float* src, float* lds_dst) {
#if __has_builtin(__builtin_amdgcn_global_load_async_to_lds_b128)
  __builtin_amdgcn_global_load_async_to_lds_b128(
      (AS1 v4i*)((AS1 const void*)src),
      (AS3 v4i*)((AS3 void*)lds_dst),
      /*imm offset*/0, /*cpol*/0);
#else
  unsigned lds_off = (unsigned)(unsigned long long)(AS3 char*)lds_dst;
  asm volatile("global_load_async_to_lds_b128 %0, %1, off"
               :: "v"(lds_off), "v"(src) : "memory");
#endif
}

__device__ __forceinline__ void wait_async0() {
#if __has_builtin(__builtin_amdgcn_s_wait_asynccnt)
  __builtin_amdgcn_s_wait_asynccnt(0);
#else
  asm volatile("s_wait_asynccnt 0" ::: "memory");
#endif
}

// ---- nonlinearities: single-trans-op v_tanh_f32 ----------------------------
__device__ __forceinline__ float tanh_fast(float x) {
#if __has_builtin(__builtin_amdgcn_tanhf)
  return __builtin_amdgcn_tanhf(x);
#else
  float e = __expf(2.0f * x);                 // v_exp_f32
  return 1.0f - 2.0f * __frcp_rn(e + 1.0f);  // v_rcp_f32
#endif
}
__device__ __forceinline__ float sigmoid_fast(float x) {
  // sigmoid(x) = 0.5 * tanh(x/2) + 0.5  -> one trans op + mul + FMA
  return __builtin_fmaf(tanh_fast(0.5f * x), 0.5f, 0.5f);
}

// ---- kernel ----------------------------------------------------------------
// One thread = one output row: 6 LSTM chains over 18 contiguous input floats,
// dotted with w_lin.  Block of 256 threads stages 256*18 floats through LDS
// via 1152 coalesced async b128 copies.
__global__ __launch_bounds__(256) void lstm_lin_kernel(
    const float* __restrict__ x,
    const float* __restrict__ w_ih, const float* __restrict__ w_hh,
    const float* __restrict__ b_ih, const float* __restrict__ b_hh,
    const float* __restrict__ w_lin, const float* __restrict__ b_lin,
    float* __restrict__ out, int n_out, long long n_x) {
  __shared__ __align__(16) float tile[256 * 18];  // 18,432 B
  const int tid = threadIdx.x;
  const long long block_base = (long long)blockIdx.x * (256LL * 18LL);
  const float* gbase = x + block_base;

  if (block_base + 4608LL <= n_x) {
    // Uniform fast path: full EXEC, no per-chunk guards.
    #pragma unroll
    for (int i = 0; i < 4; ++i) {
      const int chunk = tid + i * 256;
      async_copy_b128(gbase + chunk * 4, &tile[chunk * 4]);
    }
    if (tid < 128) {
      const int chunk = tid + 1024;
      async_copy_b128(gbase + chunk * 4, &tile[chunk * 4]);
    }
  } else {
    // Tail block (not hit for the reference sizes): per-lane guards.
    #pragma unroll
    for (int i = 0; i < 5; ++i) {
      const int chunk = tid + i * 256;
      const long long g = block_base + (long long)chunk * 4;
      if (chunk < 1152 && g + 3 < n_x) {
        async_copy_b128(x + g, &tile[chunk * 4]);
      }
    }
  }

  // Uniform weights -> scalar loads while the async copy is in flight.
  float wi[4], wh[4], bb[4];
  #pragma unroll
  for (int k = 0; k < 4; ++k) {
    wi[k] = w_ih[k];
    wh[k] = w_hh[k];
    bb[k] = b_ih[k] + b_hh[k];
  }
  float wl[18];
  #pragma unroll
  for (int k = 0; k < 18; ++k) wl[k] = w_lin[k];
  float acc = b_lin[0];

  wait_async0();
  __syncthreads();

  // Stride of 18 floats (72 B): bank = (18*lane + k) % 64 is unique per lane
  // within wave32 -> conflict-free. 8-byte aligned -> ds b64 reads.
  float a[18];
  #pragma unroll
  for (int k = 0; k < 9; ++k) {
    float2 v = *(const float2*)&tile[tid * 18 + 2 * k];
    a[2 * k]     = v.x;
    a[2 * k + 1] = v.y;
  }

  const int j = blockIdx.x * 256 + tid;

  #pragma unroll
  for (int e = 0; e < 6; ++e) {
    float h = 0.0f, c = 0.0f;
    #pragma unroll
    for (int t = 0; t < 3; ++t) {
      float xt = a[e * 3 + t];
      // gate order: i, f, g, o
      float gi = __builtin_fmaf(xt, wi[0], __builtin_fmaf(h, wh[0], bb[0]));
      float gf = __builtin_fmaf(xt, wi[1], __builtin_fmaf(h, wh[1], bb[1]));
      float gg = __builtin_fmaf(xt, wi[2], __builtin_fmaf(h, wh[2], bb[2]));
      float go = __builtin_fmaf(xt, wi[3], __builtin_fmaf(h, wh[3], bb[3]));
      float ig = sigmoid_fast(gi);
      float fg = sigmoid_fast(gf);
      float gv = tanh_fast(gg);
      float og = sigmoid_fast(go);
      c = __builtin_fmaf(fg, c, ig * gv);
      h = og * tanh_fast(c);
      acc = __builtin_fmaf(h, wl[e * 3 + t], acc);
    }
  }

  if (j < n_out) __builtin_nontemporal_store(acc, &out[j]);
}

extern "C" void kernel_launch(void* const* d_in, const int* in_sizes, int n_in,
                              void* d_out, int out_size, void* d_ws, size_t ws_size,
                              hipStream_t stream) {
  const float* x     = (const float*)d_in[0];
  const float* w_ih  = (const float*)d_in[1];
  const float* w_hh  = (const float*)d_in[2];
  const float* b_ih  = (const float*)d_in[3];
  const float* b_hh  = (const float*)d_in[4];
  const float* w_lin = (const float*)d_in[5];
  const float* b_lin = (const float*)d_in[6];
  float* out = (float*)d_out;

  long long n_x = in_sizes[0];         // 37,748,736 floats
  int n_out = (int)(n_x / 18);         // 2,097,152 outputs
  int blocks = (n_out + 255) / 256;    // 8192 blocks

  lstm_lin_kernel<<<blocks, 256, 0, stream>>>(x, w_ih, w_hh, b_ih, b_hh,
                                              w_lin, b_lin, out, n_out, n_x);
}